// TimeLSTMLayer_29910152249711
// MI455X (gfx1250) — compile-verified
//
#include <hip/hip_runtime.h>
#include <cmath>

typedef float v2f __attribute__((ext_vector_type(2)));
typedef float v8f __attribute__((ext_vector_type(8)));

constexpr int Bsz  = 128;
constexpr int Tsz  = 512;
constexpr int INs  = 64;
constexpr int OUTs = 64;
constexpr int Ks   = 8;

constexpr int KDIM   = 128;   // concat x(64) + h(64)
constexpr int JDIM   = 256;   // 4 gates * 64 outputs
constexpr int WPITCH = 132;   // padded pitch (floats) for Wt rows and xh rows
constexpr int ZPITCH = 17;    // padded pitch for zT rows (16 batch + 1 pad)

constexpr int    WT_OFF      = 0;                        // [JDIM][WPITCH]  W^T (col-major W)
constexpr int    XH_OFF      = WT_OFF + JDIM * WPITCH;   // [16][WPITCH]    [x_t | h]
constexpr int    ZT_OFF      = XH_OFF + 16 * WPITCH;     // [JDIM][ZPITCH]  z transposed
constexpr int    SMEM_FLOATS = ZT_OFF + JDIM * ZPITCH;
constexpr size_t SMEM_BYTES  = (size_t)SMEM_FLOATS * sizeof(float);

__device__ __forceinline__ float sigmoidf_(float v) {
  return 1.0f / (1.0f + expf(-v));
}

__global__ __launch_bounds__(256) void lstm_wmma_kernel(
    const float* __restrict__ x,   // (B, T, IN)
    const float* __restrict__ Wx,  // (K, 4, OUT, IN)
    const float* __restrict__ Wh,  // (K, 4, OUT, OUT)
    const float* __restrict__ wc,  // (K, 3, OUT)
    const float* __restrict__ bias,// (K, 4, OUT)
    const int*  __restrict__ nptr, // scalar n
    float* __restrict__ out) {     // (B, K, OUT)
  extern __shared__ float smem[];
  float* sWt = smem + WT_OFF;
  float* sXH = smem + XH_OFF;
  float* sZT = smem + ZT_OFF;

  const int k   = blockIdx.x;   // LSTM index 0..7
  const int bt  = blockIdx.y;   // batch tile 0..7 (16 rows each)
  const int tid = threadIdx.x;  // 0..255
  const int lane = tid & 31;
  const int wave = tid >> 5;
  const int n = nptr[0];

  // ---- load weights into LDS: sWt[j][i] with j = g*64+p, i: 0..63 = Wx row, 64..127 = Wh row
  {
    const int j = tid;                 // 256 rows, one per thread
    const int g = j >> 6, p = j & 63;
    const float* wxrow = Wx + ((size_t)((k * 4 + g) * 64 + p) * 64);
    const float* whrow = Wh + ((size_t)((k * 4 + g) * 64 + p) * 64);
    float* dst = sWt + j * WPITCH;
#pragma unroll
    for (int i = 0; i < 64; i += 4) {
      *reinterpret_cast<float4*>(dst + i)      = *reinterpret_cast<const float4*>(wxrow + i);
      *reinterpret_cast<float4*>(dst + 64 + i) = *reinterpret_cast<const float4*>(whrow + i);
    }
  }

  // ---- load x_0 tile, zero h region
  const int xrow = tid >> 4;         // 0..15
  const int xcol = (tid & 15) * 4;   // 0..60
  const size_t xbase = (size_t)(bt * 16 + xrow) * Tsz * INs + xcol;
  {
    const float4 x0 = *reinterpret_cast<const float4*>(x + xbase);  // t = 0
    *reinterpret_cast<float4*>(sXH + xrow * WPITCH + xcol) = x0;
    const float4 z4 = make_float4(0.f, 0.f, 0.f, 0.f);
    *reinterpret_cast<float4*>(sXH + xrow * WPITCH + 64 + xcol) = z4;
  }

  // ---- per-thread gate constants (fixed output channel o = tid & 63)
  const int oo = tid & 63;
  const float b_i = bias[(k * 4 + 0) * 64 + oo];
  const float b_f = bias[(k * 4 + 1) * 64 + oo];
  const float b_g = bias[(k * 4 + 2) * 64 + oo];
  const float b_o = bias[(k * 4 + 3) * 64 + oo];
  const float pwci = wc[(k * 3 + 0) * 64 + oo];
  const float pwcf = wc[(k * 3 + 1) * 64 + oo];
  const float pwco = wc[(k * 3 + 2) * 64 + oo];

  float cst[4] = {0.f, 0.f, 0.f, 0.f};  // cell state, 4 batch rows per thread
  float hst[4] = {0.f, 0.f, 0.f, 0.f};

  // ---- WMMA fragment indexing (wave32; 16x16x4 f32)
  const int j0   = wave * 32;        // this wave owns output columns [j0, j0+32)
  const int nlo  = lane & 15;
  const int kdel = (lane >> 4) * 2;  // A: lanes 0-15 -> K=kk,kk+1 ; lanes 16-31 -> K=kk+2,kk+3
  const float* arow  = sXH + nlo * WPITCH + kdel;
  const float* brow0 = sWt + (j0 + nlo) * WPITCH + kdel;
  const float* brow1 = brow0 + 16 * WPITCH;

  __syncthreads();

  for (int t = 0; t < n; ++t) {
    // prefetch next x tile into registers (hidden behind the GEMM)
    float4 xn;
    const bool more = (t + 1) < n;   // block-uniform
    if (more) xn = *reinterpret_cast<const float4*>(x + xbase + (size_t)(t + 1) * INs);

    // ---- GEMM: z(16x256) = [x_t | h](16x128) @ W^T, split accumulators to shorten dep chains
    v8f a0 = {0.f,0.f,0.f,0.f,0.f,0.f,0.f,0.f};
    v8f a1 = {0.f,0.f,0.f,0.f,0.f,0.f,0.f,0.f};
    v8f a2 = {0.f,0.f,0.f,0.f,0.f,0.f,0.f,0.f};
    v8f a3 = {0.f,0.f,0.f,0.f,0.f,0.f,0.f,0.f};
#pragma unroll
    for (int kk = 0; kk < KDIM; kk += 8) {
      const v2f af0 = *reinterpret_cast<const v2f*>(arow + kk);
      const v2f af1 = *reinterpret_cast<const v2f*>(arow + kk + 4);
      const v2f bf00 = *reinterpret_cast<const v2f*>(brow0 + kk);
      const v2f bf01 = *reinterpret_cast<const v2f*>(brow0 + kk + 4);
      const v2f bf10 = *reinterpret_cast<const v2f*>(brow1 + kk);
      const v2f bf11 = *reinterpret_cast<const v2f*>(brow1 + kk + 4);
      a0 = __builtin_amdgcn_wmma_f32_16x16x4_f32(false, af0, false, bf00, (short)0, a0, false, false);
      a1 = __builtin_amdgcn_wmma_f32_16x16x4_f32(false, af0, false, bf10, (short)0, a1, false, false);
      a2 = __builtin_amdgcn_wmma_f32_16x16x4_f32(false, af1, false, bf01, (short)0, a2, false, false);
      a3 = __builtin_amdgcn_wmma_f32_16x16x4_f32(false, af1, false, bf11, (short)0, a3, false, false);
    }

    // ---- stage z to LDS transposed: zT[j][b]  (C/D layout: VGPR r -> row r + 8*(lane>=16), col lane&15)
    {
      float* z0 = sZT + (j0 + nlo) * ZPITCH + (lane >> 4) * 8;
      float* z1 = z0 + 16 * ZPITCH;
#pragma unroll
      for (int r = 0; r < 8; ++r) {
        z0[r] = a0[r] + a2[r];
        z1[r] = a1[r] + a3[r];
      }
    }
    __syncthreads();

    // ---- gate phase: thread handles 4 batch rows at fixed output channel oo
#pragma unroll
    for (int jj = 0; jj < 4; ++jj) {
      const int br = (tid >> 6) + jj * 4;     // 0..15
      const float z_i = sZT[(0 * 64 + oo) * ZPITCH + br] + b_i;
      const float z_f = sZT[(1 * 64 + oo) * ZPITCH + br] + b_f;
      const float z_g = sZT[(2 * 64 + oo) * ZPITCH + br] + b_g;
      const float z_o = sZT[(3 * 64 + oo) * ZPITCH + br] + b_o;
      const float cp = cst[jj];
      const float ig = sigmoidf_(z_i + pwci * cp);
      const float fg = sigmoidf_(z_f + pwcf * cp);
      const float gg = tanhf(z_g);
      const float cn = fg * cp + ig * gg;
      const float og = sigmoidf_(z_o + pwco * cn);
      const float hn = og * tanhf(cn);
      cst[jj] = cn;
      hst[jj] = hn;
      sXH[br * WPITCH + 64 + oo] = hn;        // h for next step
    }
    if (more) *reinterpret_cast<float4*>(sXH + xrow * WPITCH + xcol) = xn;  // x_{t+1}
    __syncthreads();
  }

  // ---- epilogue: out[b][k][o] = h_T
#pragma unroll
  for (int jj = 0; jj < 4; ++jj) {
    const int br = (tid >> 6) + jj * 4;
    out[((size_t)(bt * 16 + br) * Ks + k) * OUTs + oo] = hst[jj];
  }
}

extern "C" void kernel_launch(void* const* d_in, const int* in_sizes, int n_in,
                              void* d_out, int out_size, void* d_ws, size_t ws_size,
                              hipStream_t stream) {
  const float* x  = (const float*)d_in[0];
  const float* Wx = (const float*)d_in[1];
  const float* Wh = (const float*)d_in[2];
  const float* wc = (const float*)d_in[3];
  const float* b  = (const float*)d_in[4];
  const int*   n  = (const int*)d_in[5];
  float* out = (float*)d_out;
  (void)in_sizes; (void)n_in; (void)out_size; (void)d_ws; (void)ws_size;

  (void)hipFuncSetAttribute(reinterpret_cast<const void*>(lstm_wmma_kernel),
                            hipFuncAttributeMaxDynamicSharedMemorySize,
                            (int)SMEM_BYTES);
  dim3 grid(Ks, Bsz / 16);   // 8 LSTMs x 8 batch tiles = 64 WGs
  lstm_wmma_kernel<<<grid, 256, SMEM_BYTES, stream>>>(x, Wx, Wh, wc, b, n, out);
}